// VQEMA_67388036874449
// MI455X (gfx1250) — compile-verified
//
#include <hip/hip_runtime.h>
#include <math.h>

typedef float v2f __attribute__((ext_vector_type(2)));
typedef float v8f __attribute__((ext_vector_type(8)));

#define NTOK  131072          // 32 * 64 * 64 tokens
#define HW    4096            // 64*64
#define CDIM  64
#define NE    1024
#define EPAD  68              // padded LDS row stride (floats) -> conflict-free b64 reads
#define TOKS_PER_BLOCK 128
#define THREADS 128           // 4 waves; each wave owns 2 x 16-token tiles
#define XQ_ELEMS  8388608     // 32*64*64*64
#define IDX_ELEMS 131072

// ---------------- ws zero: counts[0..1023] + loss accum [1024] ----------------
__global__ void vq_zero_ws(float* __restrict__ ws) {
    int i = blockIdx.x * blockDim.x + threadIdx.x;
    if (i <= NE) ws[i] = 0.0f;
}

// ---------------- main kernel ----------------
__launch_bounds__(THREADS, 1)
__global__ void vq_main(const float* __restrict__ x_in,
                        const float* __restrict__ emb,
                        float* __restrict__ xq_out,
                        float* __restrict__ idx_out,
                        float* __restrict__ ws) {
    __shared__ float emb_lds[NE * EPAD];            // 278528 B
    __shared__ float xs[TOKS_PER_BLOCK * EPAD];     //  34816 B
    __shared__ float e2_lds[NE];                    //   4096 B
    __shared__ float x2_lds[TOKS_PER_BLOCK];        //    512 B
    __shared__ int   chosen[TOKS_PER_BLOCK];        //    512 B

    const int tid  = threadIdx.x;
    const int tok0 = blockIdx.x * TOKS_PER_BLOCK;   // global token base (contiguous hw)
    const int b    = tok0 >> 12;                    // HW == 4096
    const int hw0  = tok0 & (HW - 1);

    // ---- stage codebook into LDS with float4 (row stride EPAD keeps 16B align) ----
    for (int i = tid; i < (NE * CDIM) / 4; i += THREADS) {
        const float4 v = ((const float4*)emb)[i];
        const int code = (i * 4) >> 6;
        const int k    = (i * 4) & 63;
        *(float4*)&emb_lds[code * EPAD + k] = v;
    }
    // ---- stage x tile [128 tokens][64 ch], coalesced over hw ----
    for (int i = tid; i < TOKS_PER_BLOCK * CDIM; i += THREADS) {
        const int c = i >> 7;
        const int t = i & 127;
        xs[t * EPAD + c] = x_in[(size_t)(b * CDIM + c) * HW + hw0 + t];
    }
    __syncthreads();

    // ---- per-code ||e||^2 and per-token ||x||^2 ----
    for (int code = tid; code < NE; code += THREADS) {
        float s = 0.f;
        #pragma unroll 8
        for (int k = 0; k < CDIM; ++k) { const float e = emb_lds[code * EPAD + k]; s += e * e; }
        e2_lds[code] = s;
    }
    {
        float s = 0.f;
        #pragma unroll 8
        for (int k = 0; k < CDIM; ++k) { const float x = xs[tid * EPAD + k]; s += x * x; }
        x2_lds[tid] = s;
    }
    __syncthreads();

    // ---- WMMA distance pass: each wave owns two 16-token tiles (B-fragment reuse x2) ----
    const int lane = tid & 31;
    const int wave = tid >> 5;
    const int row  = lane & 15;     // A-row (token) / B-column (code) for this lane
    const int half = lane >> 4;     // selects K pair within 16x16x4 fragment
    const int t0   = wave * 32 + row;        // tile 0 token (local)
    const int t1   = wave * 32 + 16 + row;   // tile 1 token (local)

    // preload all A fragments for both tiles: 2 x 16 x v2f (64 VGPRs)
    v2f a0[16], a1[16];
    #pragma unroll
    for (int kc = 0; kc < 16; ++kc) {
        a0[kc] = *(const v2f*)&xs[t0 * EPAD + kc * 4 + half * 2];
        a1[kc] = *(const v2f*)&xs[t1 * EPAD + kc * 4 + half * 2];
    }

    float bs0[8], bs1[8];
    int   bi0[8], bi1[8];
    #pragma unroll
    for (int v = 0; v < 8; ++v) {
        bs0[v] = 3.4e38f; bi0[v] = 0;
        bs1[v] = 3.4e38f; bi1[v] = 0;
    }

    for (int ct = 0; ct < NE / 16; ++ct) {
        const int code = ct * 16 + row;     // this lane's column's global code id
        v8f acc0 = {};
        v8f acc1 = {};
        #pragma unroll
        for (int kc = 0; kc < 16; ++kc) {
            const v2f bf = *(const v2f*)&emb_lds[code * EPAD + kc * 4 + half * 2];
            acc0 = __builtin_amdgcn_wmma_f32_16x16x4_f32(false, a0[kc], false, bf,
                                                         (short)0, acc0, false, false);
            acc1 = __builtin_amdgcn_wmma_f32_16x16x4_f32(false, a1[kc], false, bf,
                                                         (short)0, acc1, false, false);
        }
        const float e2 = e2_lds[code];
        #pragma unroll
        for (int v = 0; v < 8; ++v) {
            const float s0 = __builtin_fmaf(-2.0f, acc0[v], e2);   // ||e||^2 - 2 x.e
            if (s0 < bs0[v]) { bs0[v] = s0; bi0[v] = code; }
            const float s1 = __builtin_fmaf(-2.0f, acc1[v], e2);
            if (s1 < bs1[v]) { bs1[v] = s1; bi1[v] = code; }
        }
    }

    // cross-lane argmin within each 16-lane half (lex order -> first-min like jnp.argmin)
    #pragma unroll
    for (int off = 8; off >= 1; off >>= 1) {
        #pragma unroll
        for (int v = 0; v < 8; ++v) {
            float os = __shfl_xor(bs0[v], off, 32);
            int   oi = __shfl_xor(bi0[v], off, 32);
            if (os < bs0[v] || (os == bs0[v] && oi < bi0[v])) { bs0[v] = os; bi0[v] = oi; }
            os = __shfl_xor(bs1[v], off, 32);
            oi = __shfl_xor(bi1[v], off, 32);
            if (os < bs1[v] || (os == bs1[v] && oi < bi1[v])) { bs1[v] = os; bi1[v] = oi; }
        }
    }

    // writer lanes: lane 0 -> rows 0..7, lane 16 -> rows 8..15 (both tiles)
    if (row == 0) {
        float loss = 0.f;
        #pragma unroll
        for (int v = 0; v < 8; ++v) {
            const int r  = v + half * 8;
            // tile 0
            int tl = wave * 32 + r;
            chosen[tl] = bi0[v];
            idx_out[tok0 + tl] = (float)bi0[v];
            loss += x2_lds[tl] + bs0[v];                // ||x - e||^2
            atomicAdd(&ws[bi0[v]], 1.0f);               // histogram
            // tile 1
            tl = wave * 32 + 16 + r;
            chosen[tl] = bi1[v];
            idx_out[tok0 + tl] = (float)bi1[v];
            loss += x2_lds[tl] + bs1[v];
            atomicAdd(&ws[bi1[v]], 1.0f);
        }
        atomicAdd(&ws[NE], loss);                       // global loss accumulator
    }
    __syncthreads();

    // ---- gather + channel-first write of x_q (== forward value of x_q_out) ----
    {
        const int t  = tid;                // 0..127 -> one token per thread
        const int ci = chosen[t];
        #pragma unroll 4
        for (int c = 0; c < CDIM; ++c) {
            xq_out[(size_t)(b * CDIM + c) * HW + hw0 + t] = emb_lds[ci * EPAD + c];
        }
    }
}

// ---------------- finalize: vq_loss + perplexity ----------------
__global__ void vq_finalize(const float* __restrict__ ws, float* __restrict__ out_scalars) {
    __shared__ float red[1024];
    const int i = threadIdx.x;
    const float p = ws[i] * (1.0f / (float)NTOK);
    red[i] = p * __logf(p + 1e-10f);
    __syncthreads();
    for (int s = 512; s > 0; s >>= 1) {
        if (i < s) red[i] += red[i + s];
        __syncthreads();
    }
    if (i == 0) {
        out_scalars[0] = 0.25f * ws[NE] * (1.0f / (float)XQ_ELEMS);  // BETA * mean sq
        out_scalars[1] = __expf(-red[0]);                            // perplexity
    }
}

extern "C" void kernel_launch(void* const* d_in, const int* in_sizes, int n_in,
                              void* d_out, int out_size, void* d_ws, size_t ws_size,
                              hipStream_t stream) {
    const float* x_in = (const float*)d_in[0];
    const float* emb  = (const float*)d_in[1];
    float* out = (float*)d_out;
    float* ws  = (float*)d_ws;    // needs (NE+1)*4 bytes

    vq_zero_ws<<<5, 256, 0, stream>>>(ws);
    vq_main<<<NTOK / TOKS_PER_BLOCK, THREADS, 0, stream>>>(
        x_in, emb, out, out + XQ_ELEMS, ws);
    vq_finalize<<<1, 1024, 0, stream>>>(ws, out + XQ_ELEMS + IDX_ELEMS);
}